// S4_20461224198829
// MI455X (gfx1250) — compile-verified
//
#include <hip/hip_runtime.h>
#include <hip/hip_bf16.h>
#include <math.h>

// Problem dimensions (fixed by the reference setup)
#define BSZ  4
#define LSEQ 2048
#define DCH  768
#define NST  64
#define DN   (DCH * NST)

typedef __attribute__((ext_vector_type(2))) float v2f;
typedef __attribute__((ext_vector_type(8))) float v8f;

// ---------------------------------------------------------------------------
// Kernel 0: discretization.  dA = softplus(log_delta)*A ;
// CB = C * ((exp(dA)-1)/A) * B        (all complex, stored split re/im)
// ---------------------------------------------------------------------------
__global__ void __launch_bounds__(256)
s4_precompute(const float* __restrict__ A_real, const float* __restrict__ A_imag,
              const float* __restrict__ B_mat,  const float* __restrict__ C_mat,
              const float* __restrict__ log_delta,
              float* __restrict__ dAr, float* __restrict__ dAi,
              float* __restrict__ CBr, float* __restrict__ CBi) {
    int idx = blockIdx.x * blockDim.x + threadIdx.x;
    if (idx >= DN) return;
    int d = idx / NST;

    float delta = log1pf(expf(log_delta[d]));          // softplus
    float ar = A_real[idx], ai = A_imag[idx];
    float dr = delta * ar, di = delta * ai;            // dA

    // A_bar = exp(dA)
    float er = expf(dr);
    float s, c;
    __sincosf(di, &s, &c);
    float abr = er * c, abi = er * s;

    // (A_bar - 1) / A   (complex divide via conj)
    float nr = abr - 1.0f, ni = abi;
    float inv = 1.0f / (ar * ar + ai * ai);
    float qr = (nr * ar + ni * ai) * inv;
    float qi = (ni * ar - nr * ai) * inv;

    float b = B_mat[idx], cm = C_mat[idx];
    dAr[idx] = dr;
    dAi[idx] = di;
    CBr[idx] = cm * (qr * b);
    CBi[idx] = cm * (qi * b);
}

// ---------------------------------------------------------------------------
// Kernel 1: convolution kernel K[d,l] = Re( sum_n CB * exp(dA * l) )
//         = sum_n e^{dAr*l} * ( CBr*cos(dAi*l) - CBi*sin(dAi*l) )
// One thread per (d,l); threads in a block share d -> broadcast loads.
// ---------------------------------------------------------------------------
__global__ void __launch_bounds__(256)
s4_kernelgen(const float* __restrict__ dAr, const float* __restrict__ dAi,
             const float* __restrict__ CBr, const float* __restrict__ CBi,
             float* __restrict__ Kbuf) {
    int idx = blockIdx.x * blockDim.x + threadIdx.x;   // = d*LSEQ + l
    if (idx >= DCH * LSEQ) return;
    int   d = idx / LSEQ;
    float l = (float)(idx & (LSEQ - 1));

    const float* pr = dAr + d * NST;
    const float* pi = dAi + d * NST;
    const float* cr = CBr + d * NST;
    const float* ci = CBi + d * NST;

    float acc = 0.0f;
#pragma unroll 4
    for (int n = 0; n < NST; ++n) {
        float e = __expf(pr[n] * l);                   // decays (Re(dA) < 0)
        float s, c;
        __sincosf(pi[n] * l, &s, &c);
        acc += e * (cr[n] * c - ci[n] * s);
    }
    Kbuf[idx] = acc;
}

// ---------------------------------------------------------------------------
// Kernel 2: causal Toeplitz convolution via V_WMMA_F32_16X16X4_F32.
// One 256-thread block = 8 wave32s, all on the SAME channel d:
//   - stage the whole 8 KB K[d] row into LDS once (ds_store_b128),
//   - wave w computes output-time tile I = Ig*8 + w:
//       C(16x16) += A(16x4 Toeplitz from LDS K[d]) x B(4x16, cols = batch)
//     summed over input-time blocks J = 0..I in K-chunks of 4.
// Fragment layouts per CDNA5 ISA 7.12.2:
//   A: VGPR v, lane-group g (lane>>4): K = 2g+v, M = lane&15
//   B: VGPR v: row = 2g+v, N = lane&15
//   C/D: VGPR r: M = r + 8g, N = lane&15
// ---------------------------------------------------------------------------
__global__ void __launch_bounds__(256)
s4_conv_wmma(const float* __restrict__ u,     // (B, L, D)
             const float* __restrict__ Kbuf,  // (D, L)
             float* __restrict__ y) {         // (B, L, D)
    __shared__ float Klds[LSEQ];                       // 8 KB: full K[d] row

    const int d  = blockIdx.x >> 4;                    // 16 I-groups per channel
    const int Ig = blockIdx.x & 15;

    // ---- cooperative staged copy K[d] -> LDS, 128-bit granularity
    {
        const float4* src = (const float4*)(Kbuf + (size_t)d * LSEQ);
        float4*       dst = (float4*)Klds;
#pragma unroll
        for (int i = 0; i < (LSEQ / 4) / 256; ++i)     // 2 iters: 512 float4
            dst[threadIdx.x + i * 256] = src[threadIdx.x + i * 256];
    }
    __syncthreads();

    const int wave = threadIdx.x >> 5;
    const int lane = threadIdx.x & 31;
    const int I    = Ig * 8 + wave;                    // this wave's output block
    const int ln   = lane & 15;                        // A row (M) / B,D col (N)
    const int g    = lane >> 4;
    const int orow = I * 16 + ln;                      // absolute output time

    v8f acc = {};

    for (int J = 0; J <= I; ++J) {
        const int jbase = J * 16;

        // prefetch next J-block of u (speculative, L2)
        if (ln < BSZ && J < I) {
            const float* pf = u + ((size_t)ln * LSEQ + (size_t)(jbase + 16)) * DCH + d;
            __builtin_prefetch(pf, 0, 1);
        }

#pragma unroll
        for (int cchunk = 0; cchunk < 4; ++cchunk) {
            const int kb = jbase + cchunk * 4;         // absolute input time of chunk
            // ---- A fragment from LDS: Toeplitz K[d, orow - k], causal mask
            const int k0   = kb + 2 * g;               // K-index of a.x
            const int diff = orow - k0;
            v2f a;
            a.x = (diff >= 0) ? Klds[diff]     : 0.0f;
            a.y = (diff >= 1) ? Klds[diff - 1] : 0.0f; // k0+1 -> lag-1
            // ---- B fragment from global: u[b=ln, t, d], rows t = k0 + v
            v2f b;
            if (ln < BSZ) {
                const float* up = u + ((size_t)ln * LSEQ + (size_t)k0) * DCH + d;
                b.x = up[0];
                b.y = up[DCH];
            } else {
                b.x = 0.0f;
                b.y = 0.0f;
            }
            // (neg_a, A, neg_b, B, c_mod, C, reuse_a, reuse_b)
            acc = __builtin_amdgcn_wmma_f32_16x16x4_f32(
                false, a, false, b, (short)0, acc, false, false);
        }
    }

    // ---- store D fragment: y[b=ln, I*16 + (r + 8g), d]
#pragma unroll
    for (int r = 0; r < 8; ++r) {
        if (ln < BSZ) {
            const int mrow = r + 8 * g;
            y[((size_t)ln * LSEQ + (size_t)(I * 16 + mrow)) * DCH + d] = acc[r];
        }
    }
}

// ---------------------------------------------------------------------------
extern "C" void kernel_launch(void* const* d_in, const int* in_sizes, int n_in,
                              void* d_out, int out_size, void* d_ws, size_t ws_size,
                              hipStream_t stream) {
    const float* u         = (const float*)d_in[0];   // (4, 2048, 768)
    const float* A_real    = (const float*)d_in[1];   // (768, 64)
    const float* A_imag    = (const float*)d_in[2];   // (768, 64)
    const float* B_mat     = (const float*)d_in[3];   // (768, 64)
    const float* C_mat     = (const float*)d_in[4];   // (768, 64)
    const float* log_delta = (const float*)d_in[5];   // (768,)
    float* y = (float*)d_out;                          // (4, 2048, 768)

    float* ws   = (float*)d_ws;
    float* dAr  = ws;
    float* dAi  = ws + 1 * DN;
    float* CBr  = ws + 2 * DN;
    float* CBi  = ws + 3 * DN;
    float* Kbuf = ws + 4 * DN;                         // D*L floats (6.3 MB)

    // 0) discretize: dA, CB
    s4_precompute<<<dim3((DN + 255) / 256), dim3(256), 0, stream>>>(
        A_real, A_imag, B_mat, C_mat, log_delta, dAr, dAi, CBr, CBi);

    // 1) build K (D, L)
    s4_kernelgen<<<dim3((DCH * LSEQ + 255) / 256), dim3(256), 0, stream>>>(
        dAr, dAi, CBr, CBi, Kbuf);

    // 2) causal conv via fp32 WMMA: one block per (d, group of 8 output tiles)
    s4_conv_wmma<<<dim3(DCH * 16), dim3(256), 0, stream>>>(u, Kbuf, y);
}